// SMPConv_12214886990423
// MI455X (gfx1250) — compile-verified
//
#include <hip/hip_runtime.h>
#include <hip/hip_bf16.h>

typedef __attribute__((ext_vector_type(2))) float v2f;
typedef __attribute__((ext_vector_type(8))) float v8f;

#define KS        11
#define NPTS      4
#define PLANES    384
#define KB_STRIDE 128      // kernel bank row stride (taps padded 121 -> 128)
#define LDSW      80       // LDS tile row stride: 8 halo + 64 + 8 pad (16B-aligned rows)
#define TILE_H    74

// Wave-relative LDS byte address = low 32 bits of the generic address
// (ISA: LDS aperture addresses truncate to addr[31:0]).
static __device__ __forceinline__ unsigned int lds_off(const void* p) {
    return (unsigned int)(unsigned long long)p;
}

// ---------------------------------------------------------------------------
// Kernel 1: synthesize the 384x121 kernel bank with V_WMMA_F32_16X16X4_F32
// (M=16 channels, N=16 taps, K=4 points), plus extract the shared nonzero
// tap list (identical for all channels since the hat geometry is shared).
//
// Final kernel (after reference's transpose/flip):
//   Kf[c,u,v] = sum_p weights[c,p] * relu(1 - (|wc_y - lin[v]| + |wc_x - lin[10-u]|)/r_p)
// with lin[i] = (i-5)*0.2, tap t = u*11 + v.
// ---------------------------------------------------------------------------
__global__ __launch_bounds__(256) void smp_gen_kernel(
    const float* __restrict__ wc,       // (1,4,2)
    const float* __restrict__ radius,   // (1,4,1,1)
    const float* __restrict__ weights,  // (1,384,4)
    int*   __restrict__ tap_hdr,        // [0]=ntaps, [1..128]=tap indices
    float* __restrict__ kbank)          // (384,128)
{
    __shared__ float sh_flags[KB_STRIDE];

    const int tid  = threadIdx.x;
    const int wave = tid >> 5;          // 0..7  -> N-tile (tap tile)
    const int lane = tid & 31;
    const int half = lane >> 4;         // 0/1   -> K pair selector
    const int lid  = lane & 15;

    // ---- B fragment: hat basis values for this wave's 16 taps -------------
    const int t = (wave << 4) + lid;    // global tap id 0..127
    const int u = t / KS;
    const int v = t - u * KS;
    const float gy = (float)(v - 5) * 0.2f;   // lin[v]
    const float gx = (float)(5 - u) * 0.2f;   // lin[10-u]

    float hat[NPTS];
    float hsum = 0.0f;
#pragma unroll
    for (int p = 0; p < NPTS; ++p) {
        float dy = fabsf(wc[2 * p + 0] - gy);
        float dx = fabsf(wc[2 * p + 1] - gx);
        float h  = fmaxf(1.0f - (dy + dx) / radius[p], 0.0f);
        hat[p]   = (t < KS * KS) ? h : 0.0f;
        hsum    += hat[p];
    }

    v2f bfrag;
    bfrag.x = hat[2 * half + 0];        // K = 2*half
    bfrag.y = hat[2 * half + 1];        // K = 2*half + 1

    // ---- 24 M-tiles of 16 channels each; B fragment reused ----------------
    for (int mt = 0; mt < PLANES / 16; ++mt) {
        const int c = (mt << 4) + lid;
        v2f afrag;
        afrag.x = weights[c * NPTS + 2 * half + 0];
        afrag.y = weights[c * NPTS + 2 * half + 1];

        v8f acc = {};
        acc = __builtin_amdgcn_wmma_f32_16x16x4_f32(
            /*neg_a=*/false, afrag, /*neg_b=*/false, bfrag,
            /*c_mod=*/(short)0, acc, /*reuse_a=*/false, /*reuse_b=*/false);

#pragma unroll
        for (int vv = 0; vv < 8; ++vv) {
            const int row = (mt << 4) + vv + 8 * half;   // channel
            const int col = (wave << 4) + lid;           // tap
            kbank[row * KB_STRIDE + col] = acc[vv];
        }
    }

    // ---- shared nonzero-tap mask + compaction -----------------------------
    if (half == 0 && t < KB_STRIDE) sh_flags[t] = hsum;
    __syncthreads();

    if (tid == 0) {
        int n = 0;
        for (int i = 0; i < KS * KS; ++i)
            if (sh_flags[i] > 0.0f) tap_hdr[1 + n++] = i;
        tap_hdr[0] = n;
    }
}

// ---------------------------------------------------------------------------
// Kernel 2: depthwise conv. One workgroup per (n, c) plane. The haloed
// 74x80 tile is staged with CDNA5 async B128 loads straight into LDS
// (ASYNCcnt path); compute is register-blocked: 16 consecutive output rows
// per thread, tap loop outermost, so each tap costs 1 coef + 1 delta +
// 16 immediate-offset tile ds_loads for 16 FMAs.
// ---------------------------------------------------------------------------
__global__ __launch_bounds__(256) void smp_conv_kernel(
    const float* __restrict__ x,        // (16,384,64,64)
    const int*   __restrict__ tap_hdr,
    const float* __restrict__ kbank,
    float* __restrict__ out)            // (16,384,64,64)
{
    __shared__ float sh_tile[TILE_H * LDSW];
    __shared__ float sh_coef[KB_STRIDE];
    __shared__ int   sh_delta[KB_STRIDE];

    const int blk = blockIdx.x;
    const int c   = blk % PLANES;
    const int n   = blk / PLANES;
    const int tid = threadIdx.x;

    const int ntaps = tap_hdr[0];       // uniform -> scalar load

    if (tid < ntaps) {
        const int t  = tap_hdr[1 + tid];
        const int u  = t / KS;
        const int v  = t - u * KS;
        sh_delta[tid] = u * LDSW + v;
        sh_coef[tid]  = kbank[c * KB_STRIDE + t];
    }

    // ---- phase 1: zero the whole tile (covers halo) -----------------------
    for (int i = tid; i < TILE_H * LDSW; i += 256)
        sh_tile[i] = 0.0f;
    __syncthreads();

    // ---- phase 2: async-stage the 64x64 interior, 16B granules ------------
    // image row gy -> tile row gy+5; image col gx -> tile col gx+8.
    const float* __restrict__ xp = x + (size_t)(n * PLANES + c) * 4096;

#pragma unroll
    for (int k = 0; k < 4; ++k) {
        const int idx = tid + (k << 8);          // 0..1023
        const int gy  = idx >> 4;                // 0..63
        const int gx  = (idx & 15) << 2;         // 0,4,...,60
        const float* gsrc = xp + (gy << 6) + gx;
        const unsigned int ldst =
            lds_off(&sh_tile[(gy + 5) * LDSW + 8 + gx]);
        asm volatile("global_load_async_to_lds_b128 %0, %1, off"
                     :: "v"(ldst), "v"(gsrc) : "memory");
    }
    asm volatile("s_wait_asynccnt 0x0" ::: "memory");
    __syncthreads();

    // ---- phase 3: register-blocked compute --------------------------------
    const int xcol  = tid & 63;          // = w
    const int yrow0 = (tid >> 6) << 4;   // 0,16,32,48: 16 consecutive rows each
    const int base0 = yrow0 * LDSW + xcol + 3;   // + delta(u,v) + r*LDSW

    float acc[16];
#pragma unroll
    for (int r = 0; r < 16; ++r) acc[r] = 0.0f;

    for (int j = 0; j < ntaps; ++j) {
        const float cf = sh_coef[j];
        const int   dl = sh_delta[j];
        const float* __restrict__ tp = &sh_tile[base0 + dl];
#pragma unroll
        for (int r = 0; r < 16; ++r)
            acc[r] = fmaf(cf, tp[r * LDSW], acc[r]);
    }

    float* __restrict__ op = out + (size_t)(n * PLANES + c) * 4096 + xcol;
#pragma unroll
    for (int r = 0; r < 16; ++r)
        op[(yrow0 + r) << 6] = acc[r];
}

// ---------------------------------------------------------------------------
extern "C" void kernel_launch(void* const* d_in, const int* in_sizes, int n_in,
                              void* d_out, int out_size, void* d_ws, size_t ws_size,
                              hipStream_t stream)
{
    const float* x       = (const float*)d_in[0];
    const float* wc      = (const float*)d_in[1];
    const float* radius  = (const float*)d_in[2];
    const float* weights = (const float*)d_in[3];

    int*   tap_hdr = (int*)d_ws;                          // 129 ints
    float* kbank   = (float*)((char*)d_ws + 1024);        // 384*128 floats

    smp_gen_kernel<<<1, 256, 0, stream>>>(wc, radius, weights, tap_hdr, kbank);
    smp_conv_kernel<<<16 * PLANES, 256, 0, stream>>>(x, tap_hdr, kbank,
                                                     (float*)d_out);
}